// GRUTopK_2637109920178
// MI455X (gfx1250) — compile-verified
//
#include <hip/hip_runtime.h>
#include <math.h>

// ---------------------------------------------------------------------------
// GRU-TopK for MI455X (gfx1250, wave32).
// I=256, O=2048, R=200000.
// Stage 1: scores = node_embs@p / ||p|| + mask   (204.8 MB stream -> HBM bound)
// Stage 2: radix-select top-2048 + bitonic sort (value desc, index asc)
// Stage 3: gather xT[j][i] = node_embs[idx[j]][i] * tanh(val[j])  (transposed)
// Stage 4: fused GRU GEMMs via V_WMMA_F32_16X16X4_F32 (fp32-exact tensor path)
// ---------------------------------------------------------------------------

#define I_DIM 256
#define O_DIM 2048
#define R_DIM 200000
#define K_SEL 2048

typedef __attribute__((ext_vector_type(2))) float v2f;
typedef __attribute__((ext_vector_type(8))) float v8f;

__device__ __forceinline__ v8f wmma_f32x4(v2f a, v2f b, v8f c) {
  // D = A(16x4 f32) * B(4x16 f32) + C(16x16 f32)
  return __builtin_amdgcn_wmma_f32_16x16x4_f32(
      /*neg_a=*/false, a, /*neg_b=*/false, b,
      /*c_mod=*/(short)0, c, /*reuse_a=*/false, /*reuse_b=*/false);
}

__device__ __forceinline__ unsigned f2key(float f) {
  unsigned u = __float_as_uint(f);
  return (u & 0x80000000u) ? ~u : (u | 0x80000000u);  // monotone ascending
}
__device__ __forceinline__ float key2f(unsigned k) {
  unsigned u = (k & 0x80000000u) ? (k ^ 0x80000000u) : ~k;
  return __uint_as_float(u);
}
__device__ __forceinline__ float sigmoidf(float x) {
  return 1.0f / (1.0f + expf(-x));
}

// -------------------- Stage 0: ||p||, state init, hist zero ----------------
__global__ __launch_bounds__(256) void init_kernel(const float* __restrict__ p,
                                                   unsigned* state, float* stateF,
                                                   unsigned* hist) {
  __shared__ float red[256];
  int t = threadIdx.x;
  float v = p[t];
  red[t] = v * v;
  hist[t] = 0u;
  __syncthreads();
  for (int s = 128; s > 0; s >>= 1) {
    if (t < s) red[t] += red[t + s];
    __syncthreads();
  }
  if (t == 0) {
    stateF[4] = rsqrtf(red[0]);  // 1/||p||
    state[0] = 0u;               // radix prefix
    state[1] = K_SEL;            // rank needed within matching subset
    state[2] = 0u;               // compaction counter (>T)
    state[3] = 0u;               // compaction counter (==T)
  }
}

// -------------------- Stage 1: scores -> sortable keys ---------------------
// One wave32 per row; lane reads 8 contiguous floats (2x float4).
__global__ __launch_bounds__(256) void score_kernel(
    const float* __restrict__ ne, const float* __restrict__ p,
    const float* __restrict__ mask, const float* __restrict__ stateF,
    unsigned* __restrict__ keys) {
  int lane = threadIdx.x & 31;
  int r = blockIdx.x * 8 + (threadIdx.x >> 5);
  const float4* row = (const float4*)(ne + (size_t)r * I_DIM + lane * 8);
  const float4* pv = (const float4*)(p + lane * 8);
  float4 a0 = row[0], a1 = row[1];
  float4 p0 = pv[0], p1 = pv[1];
  float sum = a0.x * p0.x + a0.y * p0.y + a0.z * p0.z + a0.w * p0.w +
              a1.x * p1.x + a1.y * p1.y + a1.z * p1.z + a1.w * p1.w;
  sum += __shfl_down(sum, 16);
  sum += __shfl_down(sum, 8);
  sum += __shfl_down(sum, 4);
  sum += __shfl_down(sum, 2);
  sum += __shfl_down(sum, 1);
  if (lane == 0) {
    float s = sum * stateF[4] + mask[r];
    keys[r] = f2key(s);
  }
}

// -------------------- Stage 2a: radix histogram (one digit) ----------------
__global__ __launch_bounds__(256) void hist_kernel(const unsigned* __restrict__ keys,
                                                   const unsigned* __restrict__ state,
                                                   unsigned* __restrict__ hist,
                                                   int shift) {
  __shared__ unsigned lh[256];
  lh[threadIdx.x] = 0u;
  __syncthreads();
  unsigned prefix = state[0];
  unsigned maskHi = (shift == 24) ? 0u : (0xFFFFFFFFu << (shift + 8));
  for (int r = blockIdx.x * blockDim.x + threadIdx.x; r < R_DIM;
       r += gridDim.x * blockDim.x) {
    unsigned k = keys[r];
    if ((k & maskHi) == prefix) atomicAdd(&lh[(k >> shift) & 255u], 1u);
  }
  __syncthreads();
  if (lh[threadIdx.x]) atomicAdd(&hist[threadIdx.x], lh[threadIdx.x]);
}

// -------------------- Stage 2b: pick digit, re-zero hist -------------------
__global__ void select_kernel(unsigned* hist, unsigned* state, int shift) {
  unsigned kneed = state[1], cum = 0;
  for (int d = 255; d >= 0; --d) {
    unsigned c = hist[d];
    if (cum + c >= kneed) {
      state[0] |= ((unsigned)d) << shift;
      state[1] = kneed - cum;
      break;
    }
    cum += c;
  }
  for (int i = 0; i < 256; ++i) hist[i] = 0u;  // ready for next pass
}

// -------------------- Stage 2c: compact candidates -------------------------
__global__ __launch_bounds__(256) void compact_kernel(
    const unsigned* __restrict__ keys, unsigned* state,
    unsigned* __restrict__ ckey, unsigned* __restrict__ cidx) {
  int r = blockIdx.x * blockDim.x + threadIdx.x;
  if (r >= R_DIM) return;
  unsigned T = state[0];
  unsigned kneed = state[1];  // how many ==T to take
  unsigned n_gt = K_SEL - kneed;
  unsigned k = keys[r];
  if (k > T) {
    unsigned pos = atomicAdd(&state[2], 1u);
    if (pos < (unsigned)K_SEL) { ckey[pos] = k; cidx[pos] = (unsigned)r; }
  } else if (k == T) {
    unsigned t = atomicAdd(&state[3], 1u);
    if (t < kneed) { ckey[n_gt + t] = k; cidx[n_gt + t] = (unsigned)r; }
  }
}

// -------------------- Stage 2d: bitonic sort 2048 (desc key, asc idx) ------
__global__ __launch_bounds__(1024) void sort_kernel(
    const unsigned* __restrict__ ckey, const unsigned* __restrict__ cidx,
    float* __restrict__ vals, int* __restrict__ idxOut) {
  __shared__ unsigned long long s[K_SEL];
  int tid = threadIdx.x;
  for (int j = tid; j < K_SEL; j += 1024)
    s[j] = ((unsigned long long)ckey[j] << 32) | (unsigned)(~cidx[j]);
  __syncthreads();
  for (int size = 2; size <= K_SEL; size <<= 1) {
    for (int stride = size >> 1; stride > 0; stride >>= 1) {
      int t = tid;  // exactly N/2 = 1024 compare-exchanges per step
      int lo = 2 * t - (t & (stride - 1));
      int hi = lo + stride;
      bool desc = ((lo & size) == 0);
      unsigned long long a = s[lo], b = s[hi];
      bool sw = desc ? (a < b) : (a > b);
      if (sw) { s[lo] = b; s[hi] = a; }
      __syncthreads();
    }
  }
  for (int j = tid; j < K_SEL; j += 1024) {
    unsigned key = (unsigned)(s[j] >> 32);
    unsigned id = ~((unsigned)(s[j] & 0xFFFFFFFFull));
    vals[j] = key2f(key);
    idxOut[j] = (int)id;
  }
}

// -------------------- Stage 3: gather transposed activations ---------------
// xT[j][i] = node_embs[idx[j]][i] * tanh(vals[j])  (2048 x 256, row-major)
__global__ __launch_bounds__(256) void gather_kernel(
    const float* __restrict__ ne, const float* __restrict__ vals,
    const int* __restrict__ idx, float* __restrict__ xT) {
  int j = blockIdx.x;
  int i = threadIdx.x;
  float t = tanhf(vals[j]);
  xT[j * I_DIM + i] = ne[(size_t)idx[j] * I_DIM + i] * t;
}

// -------------------- Stage 4a: update/reset gates (4 fused WMMA GEMMs) ----
// Wave per 16x16 tile of the (256 x 2048) outputs. K=256 in steps of 4.
__global__ __launch_bounds__(256) void gemm_ur_kernel(
    const float* __restrict__ Wu, const float* __restrict__ Uu,
    const float* __restrict__ bu, const float* __restrict__ Wr,
    const float* __restrict__ Ur, const float* __restrict__ br,
    const float* __restrict__ xT, const float* __restrict__ h,
    float* __restrict__ upd, float* __restrict__ rh) {
  int wave = threadIdx.x >> 5;
  int lane = threadIdx.x & 31;
  int tile = blockIdx.x * 8 + wave;   // 2048 tiles = 16 (M) x 128 (N)
  int tm = tile & 15;
  int tn = tile >> 4;
  int half = lane >> 4;
  int lm = lane & 15;
  int m = tm * 16 + lm;               // A-fragment row
  int n = tn * 16 + lm;               // B-fragment / output column
  const float* wuR = Wu + m * I_DIM;
  const float* uuR = Uu + m * I_DIM;
  const float* wrR = Wr + m * I_DIM;
  const float* urR = Ur + m * I_DIM;
  const float* xR = xT + n * I_DIM;
  v8f cu = {}, cr = {};
  for (int k0 = 0; k0 < I_DIM; k0 += 4) {
    int kb = k0 + 2 * half;           // lane-half K split per ISA A/B layout
    v2f awu = *(const v2f*)(wuR + kb);
    v2f auu = *(const v2f*)(uuR + kb);
    v2f awr = *(const v2f*)(wrR + kb);
    v2f aur = *(const v2f*)(urR + kb);
    v2f bx = *(const v2f*)(xR + kb);
    v2f bh;
    bh.x = h[kb * O_DIM + n];
    bh.y = h[(kb + 1) * O_DIM + n];
    cu = wmma_f32x4(awu, bx, cu);
    cu = wmma_f32x4(auu, bh, cu);
    cr = wmma_f32x4(awr, bx, cr);
    cr = wmma_f32x4(aur, bh, cr);
  }
#pragma unroll
  for (int v = 0; v < 8; ++v) {
    int md = tm * 16 + v + 8 * half;  // C/D layout: VGPR v, lane-half
    int o = md * O_DIM + n;
    float u = sigmoidf(cu[v] + bu[o]);
    float r = sigmoidf(cr[v] + br[o]);
    upd[o] = u;
    rh[o] = r * h[o];
  }
}

// -------------------- Stage 4b: candidate state + final blend --------------
__global__ __launch_bounds__(256) void gemm_h_kernel(
    const float* __restrict__ Wh, const float* __restrict__ Uh,
    const float* __restrict__ bh, const float* __restrict__ xT,
    const float* __restrict__ rh, const float* __restrict__ upd,
    const float* __restrict__ h, float* __restrict__ out) {
  int wave = threadIdx.x >> 5;
  int lane = threadIdx.x & 31;
  int tile = blockIdx.x * 8 + wave;
  int tm = tile & 15;
  int tn = tile >> 4;
  int half = lane >> 4;
  int lm = lane & 15;
  int m = tm * 16 + lm;
  int n = tn * 16 + lm;
  const float* whR = Wh + m * I_DIM;
  const float* uhR = Uh + m * I_DIM;
  const float* xR = xT + n * I_DIM;
  v8f ch = {};
  for (int k0 = 0; k0 < I_DIM; k0 += 4) {
    int kb = k0 + 2 * half;
    v2f awh = *(const v2f*)(whR + kb);
    v2f auh = *(const v2f*)(uhR + kb);
    v2f bx = *(const v2f*)(xR + kb);
    v2f bv;
    bv.x = rh[kb * O_DIM + n];
    bv.y = rh[(kb + 1) * O_DIM + n];
    ch = wmma_f32x4(awh, bx, ch);
    ch = wmma_f32x4(auh, bv, ch);
  }
#pragma unroll
  for (int v = 0; v < 8; ++v) {
    int md = tm * 16 + v + 8 * half;
    int o = md * O_DIM + n;
    float hcap = tanhf(ch[v] + bh[o]);
    float u = upd[o];
    out[o] = (1.0f - u) * h[o] + u * hcap;
  }
}

// ---------------------------------------------------------------------------
extern "C" void kernel_launch(void* const* d_in, const int* in_sizes, int n_in,
                              void* d_out, int out_size, void* d_ws, size_t ws_size,
                              hipStream_t stream) {
  const float* node_embs = (const float*)d_in[0];
  const float* history   = (const float*)d_in[1];
  const float* mask      = (const float*)d_in[2];
  const float* p         = (const float*)d_in[3];
  const float* Wu        = (const float*)d_in[4];
  const float* Uu        = (const float*)d_in[5];
  const float* bu        = (const float*)d_in[6];
  const float* Wr        = (const float*)d_in[7];
  const float* Ur        = (const float*)d_in[8];
  const float* br        = (const float*)d_in[9];
  const float* Wh        = (const float*)d_in[10];
  const float* Uh        = (const float*)d_in[11];
  const float* bh        = (const float*)d_in[12];
  float* out = (float*)d_out;

  // Workspace layout (bytes, 256-aligned sections)
  char* ws = (char*)d_ws;
  unsigned* keys  = (unsigned*)(ws + 0);          // 200000 u32 = 800000 B
  unsigned* hist  = (unsigned*)(ws + 800000);     // 256 u32
  unsigned* state = (unsigned*)(ws + 801024);     // 8 u32 (prefix,kneed,cnts,rnorm)
  float* stateF   = (float*)(ws + 801024);
  unsigned* ckey  = (unsigned*)(ws + 801280);     // 2048 u32
  unsigned* cidx  = (unsigned*)(ws + 809472);     // 2048 u32
  float* vals     = (float*)(ws + 817664);        // 2048 f32
  int* idx        = (int*)(ws + 825856);          // 2048 i32
  float* xT       = (float*)(ws + 834048);        // 2048x256 f32 = 2 MB
  float* upd      = (float*)(ws + 2931200);       // 256x2048 f32 = 2 MB
  float* rh       = (float*)(ws + 5028352);       // 256x2048 f32 = 2 MB

  // Stage 0: 1/||p||, state + histogram init
  init_kernel<<<1, 256, 0, stream>>>(p, state, stateF, hist);
  // Stage 1: scores (8 rows per 256-thread block; 200000/8 = 25000 blocks)
  score_kernel<<<25000, 256, 0, stream>>>(node_embs, p, mask, stateF, keys);
  // Stage 2: 4-pass radix select of the 2048th largest key
  for (int shift = 24; shift >= 0; shift -= 8) {
    hist_kernel<<<240, 256, 0, stream>>>(keys, state, hist, shift);
    select_kernel<<<1, 1, 0, stream>>>(hist, state, shift);
  }
  compact_kernel<<<(R_DIM + 255) / 256, 256, 0, stream>>>(keys, state, ckey, cidx);
  sort_kernel<<<1, 1024, 0, stream>>>(ckey, cidx, vals, idx);
  // Stage 3: gather scaled top-k rows (transposed for WMMA B-fragments)
  gather_kernel<<<K_SEL, 256, 0, stream>>>(node_embs, vals, idx, xT);
  // Stage 4: fused GRU GEMMs on the WMMA path (2048 tiles, 8 waves/block)
  gemm_ur_kernel<<<256, 256, 0, stream>>>(Wu, Uu, bu, Wr, Ur, br, xT, history,
                                          upd, rh);
  gemm_h_kernel<<<256, 256, 0, stream>>>(Wh, Uh, bh, xT, rh, upd, history, out);
}